// P2MLoss_59459527246412
// MI455X (gfx1250) — compile-verified
//
#include <hip/hip_runtime.h>

// ---------------------------------------------------------------------------
// P2M loss for MI455X (gfx1250, wave32).
// Chamfer distance tiles computed with V_WMMA_F32_16X16X4_F32:
//   D[m][n] = |a_m|^2 + |b_n|^2 - 2 a.b
// A row  = (ax, ay, az, 1)           (16x4 f32 A fragment, 2 VGPRs/lane)
// B col  = (-2bx, -2by, -2bz, |b|^2) (4x16 f32 B fragment, 2 VGPRs/lane)
// C init = |a_m|^2 broadcast per row (8 VGPR f32 accumulator)
// Each wave owns RPW=4 row tiles -> 4 WMMAs per staged B fragment.
// rest[] is staged once per block into LDS, pre-scaled & padded.
// ---------------------------------------------------------------------------

typedef __attribute__((ext_vector_type(2))) float v2f;
typedef __attribute__((ext_vector_type(8))) float v8f;

#define B_          8
#define NGT         10000
#define ROW_TILES   625      // NGT / 16 exactly
#define WPB         8        // waves per block
#define RPW         4        // row tiles per wave
#define INF_F       3.402823e38f
#define MAXCOLS     2480     // 155 col tiles * 16

// Float atomic-min via monotonic int trick (valid for mixed signs when the
// destination is initialized to +INF). Works for LDS and global pointers.
__device__ inline void atomicMinF(float* addr, float val) {
    if (val >= 0.0f) atomicMin((int*)addr, __float_as_int(val));
    else             atomicMax((unsigned int*)addr, __float_as_uint(val));
}

__device__ inline float waveSum(float v) {
#pragma unroll
    for (int off = 16; off >= 1; off >>= 1) v += __shfl_xor(v, off, 32);
    return v;
}

// ---------------------------------------------------------------------------
__global__ void init_kernel(float* out, float* d2, int nd2) {
    int i = blockIdx.x * blockDim.x + threadIdx.x;
    if (i < 4) out[i] = 0.0f;
    for (int j = i; j < nd2; j += gridDim.x * blockDim.x) d2[j] = INF_F;
}

// ---------------------------------------------------------------------------
// One wave = RPW consecutive 16-row tiles of gt; loops over all 16-col tiles
// of rest (staged in LDS). Block = 8 waves sharing an LDS column-min array.
__global__ __launch_bounds__(256) void chamfer_kernel(
        const float* __restrict__ gt, const float* __restrict__ rest,
        float* __restrict__ d2buf, float* __restrict__ out,
        int npts, int ntiles) {
    __shared__ float  colmin[MAXCOLS];
    __shared__ float4 restS[MAXCOLS];   // (-2x, -2y, -2z, |b|^2), padded

    const int tid  = threadIdx.x;
    const int b    = blockIdx.y;
    const int wid  = tid >> 5;
    const int lane = tid & 31;
    const int half = lane >> 4;   // 0: lanes 0-15, 1: lanes 16-31
    const int m16  = lane & 15;
    const int ncols = ntiles * 16;

    // ---- stage rest points (pre-scaled) + init column mins ----
    {
        const float* rb = rest + (size_t)b * npts * 3;
        for (int i = tid; i < ncols; i += 256) {
            float rx, ry, rz;
            if (i < npts) { rx = rb[i*3+0]; ry = rb[i*3+1]; rz = rb[i*3+2]; }
            else          { rx = 1.0e18f;   ry = 1.0e18f;   rz = 1.0e18f; } // pad -> huge dist
            restS[i]  = make_float4(-2.0f*rx, -2.0f*ry, -2.0f*rz,
                                    rx*rx + ry*ry + rz*rz);
            colmin[i] = INF_F;
        }
    }
    __syncthreads();

    // ---- per-sub-tile A fragments / C accumulators (wave-uniform activity) ----
    const int rt0 = (blockIdx.x * WPB + wid) * RPW;
    v2f  a[RPW];
    v8f  c[RPW];
    v8f  rmin[RPW];
    bool act[RPW];
#pragma unroll
    for (int r = 0; r < RPW; ++r) {
        const int rt = rt0 + r;
        act[r] = (rt < ROW_TILES);
        if (act[r]) {
            const float* g = gt + ((size_t)b * NGT + (size_t)rt * 16 + m16) * 3;
            float gx = g[0], gy = g[1], gz = g[2];
            a[r].x = half ? gz   : gx;              // K = 2*half
            a[r].y = half ? 1.0f : gy;              // K = 2*half + 1 (K3 == 1)
            float np  = half ? gz * gz : (gx * gx + gy * gy);
            float nrm = np + __shfl_xor(np, 16, 32);
#pragma unroll
            for (int j = 0; j < 8; ++j) c[r][j] = __shfl(nrm, j + 8 * half, 32);
        } else {
            // Inert sub-tile: A=0, C=+INF -> D = INF + 0 = INF (no NaN),
            // harmless for col-min; excluded from row sum by act[].
            a[r].x = 0.0f; a[r].y = 0.0f;
#pragma unroll
            for (int j = 0; j < 8; ++j) c[r][j] = INF_F;
        }
#pragma unroll
        for (int j = 0; j < 8; ++j) rmin[r][j] = INF_F;
    }

    // ---- hot loop: one LDS B fragment -> RPW WMMAs ----
    for (int ct = 0; ct < ntiles; ++ct) {
        const int n = ct * 16 + m16;
        float4 e = restS[n];
        v2f bf;
        bf.x = half ? e.z : e.x;
        bf.y = half ? e.w : e.y;

        float cm = INF_F;
#pragma unroll
        for (int r = 0; r < RPW; ++r) {
            v8f d = __builtin_amdgcn_wmma_f32_16x16x4_f32(
                        false, a[r], false, bf, (short)0, c[r], false, false);
#pragma unroll
            for (int j = 0; j < 8; ++j) {
                rmin[r][j] = fminf(rmin[r][j], d[j]);
                cm         = fminf(cm, d[j]);
            }
        }
        cm = fminf(cm, __shfl_xor(cm, 16, 32));   // min over all rows, both halves
        if (half == 0) atomicMinF(&colmin[n], cm);
    }

    // ---- row-min reduce + d1 partial sum ----
    float s = 0.0f;
#pragma unroll
    for (int r = 0; r < RPW; ++r) {
        if (act[r]) {                              // wave-uniform branch
#pragma unroll
            for (int off = 1; off < 16; off <<= 1) {
#pragma unroll
                for (int j = 0; j < 8; ++j)
                    rmin[r][j] = fminf(rmin[r][j], __shfl_xor(rmin[r][j], off, 32));
            }
            float t = 0.0f;
#pragma unroll
            for (int j = 0; j < 8; ++j) t += rmin[r][j]; // lane0: rows 0-7, lane16: rows 8-15
            s += t;
        }
    }
    s += __shfl_xor(s, 16, 32);
    if (lane == 0 && rt0 < ROW_TILES)
        atomicAdd(out + 3, s * (1.0f / (float)(B_ * NGT)));

    __syncthreads();
    for (int i = tid; i < npts; i += 256)
        atomicMinF(&d2buf[(size_t)b * npts + i], colmin[i]);
}

// ---------------------------------------------------------------------------
__global__ void d2_reduce_kernel(const float* __restrict__ d2buf,
                                 float* __restrict__ out, int n, float scale) {
    int i = blockIdx.x * blockDim.x + threadIdx.x;
    float v = 0.0f;
    for (int j = i; j < n; j += gridDim.x * blockDim.x) v += d2buf[j];
    v = waveSum(v);
    if ((threadIdx.x & 31) == 0) atomicAdd(out + 3, v * scale);
}

// ---------------------------------------------------------------------------
__global__ void edge_kernel(const float* __restrict__ pred,
                            const int* __restrict__ edges,
                            float* __restrict__ out,
                            int npts, int nedges, float scale) {
    int i = blockIdx.x * blockDim.x + threadIdx.x;
    int total = B_ * nedges;
    float v = 0.0f;
    if (i < total) {
        int b = i / nedges, k = i - b * nedges;
        int e0 = edges[2 * k], e1 = edges[2 * k + 1];
        const float* p = pred + (size_t)b * npts * 3;
        float dx = p[e0*3+0] - p[e1*3+0];
        float dy = p[e0*3+1] - p[e1*3+1];
        float dz = p[e0*3+2] - p[e1*3+2];
        v = dx*dx + dy*dy + dz*dz;
    }
    v = waveSum(v);
    if ((threadIdx.x & 31) == 0) atomicAdd(out + 0, v * scale);
}

// ---------------------------------------------------------------------------
// lap(x) = x - sum(valid nbrs)/cnt ; (l1-l2) uses identical idx/cnt, so
// (l1-l2) = (before-pred) - sum(before_nb - pred_nb)/cnt.
__global__ void lap_kernel(const float* __restrict__ pred,
                           const float* __restrict__ before,
                           const int* __restrict__ lap_idx,
                           float* __restrict__ out,
                           int npts, float lapw, float movew) {
    int i = blockIdx.x * blockDim.x + threadIdx.x;
    int total = B_ * npts;
    float lv = 0.0f, mv = 0.0f;
    if (i < total) {
        int b = i / npts, v = i - b * npts;
        const float* p = pred   + ((size_t)b * npts + v) * 3;
        const float* q = before + ((size_t)b * npts + v) * 3;
        float dx = q[0] - p[0], dy = q[1] - p[1], dz = q[2] - p[2];
        float sx = 0.0f, sy = 0.0f, sz = 0.0f;
        const int* row = lap_idx + v * 10;   // cols 0-7 nbrs, 8 self, 9 cnt
        int cnt = row[9];
#pragma unroll
        for (int j = 0; j < 8; ++j) {
            int nb = row[j];
            if (nb >= 0) {
                const float* pn = pred   + ((size_t)b * npts + nb) * 3;
                const float* qn = before + ((size_t)b * npts + nb) * 3;
                sx += qn[0] - pn[0]; sy += qn[1] - pn[1]; sz += qn[2] - pn[2];
            }
        }
        float inv = 1.0f / (float)cnt;
        float lx = dx - sx * inv, ly = dy - sy * inv, lz = dz - sz * inv;
        lv = lx*lx + ly*ly + lz*lz;
        mv = dx*dx + dy*dy + dz*dz;
    }
    lv = waveSum(lv);
    mv = waveSum(mv);
    if ((threadIdx.x & 31) == 0) {
        atomicAdd(out + 1, lv * lapw);
        if (movew != 0.0f) atomicAdd(out + 2, mv * movew);
    }
}

// ---------------------------------------------------------------------------
extern "C" void kernel_launch(void* const* d_in, const int* in_sizes, int n_in,
                              void* d_out, int out_size, void* d_ws, size_t ws_size,
                              hipStream_t stream) {
    // setup_inputs() dict order:
    // 0: gt_coord
    // per level i (i=0,1,2), base = 1 + 5*i:
    //   +0 pred_coord, +1 pred_before, +2 rest_coord, +3 lap_idx, +4 edges
    const float* gt = (const float*)d_in[0];
    const float* pred[3]   = {(const float*)d_in[1],  (const float*)d_in[6],  (const float*)d_in[11]};
    const float* before[3] = {(const float*)d_in[2],  (const float*)d_in[7],  (const float*)d_in[12]};
    const float* rest[3]   = {(const float*)d_in[3],  (const float*)d_in[8],  (const float*)d_in[13]};
    const int*   lapi[3]   = {(const int*)d_in[4],    (const int*)d_in[9],    (const int*)d_in[14]};
    const int*   edges[3]  = {(const int*)d_in[5],    (const int*)d_in[10],   (const int*)d_in[15]};
    float* out = (float*)d_out;   // [edge, lap, move, chamfer]
    float* ws  = (float*)d_ws;    // d2 min buffers, ~104 KB

    const int   npts[3]   = {156, 618, 2466};
    const int   nedges[3] = {462, 1848, 7392};
    const float lapc[3]   = {0.2f, 1.0f, 1.0f};
    const int   ntiles[3] = {10, 39, 155};

    int d2off[3]; int nd2 = 0;
    for (int i = 0; i < 3; ++i) { d2off[i] = nd2; nd2 += B_ * npts[i]; }

    init_kernel<<<dim3(102), dim3(256), 0, stream>>>(out, ws, nd2);

    const int tiles_per_block = WPB * RPW;                       // 32
    for (int i = 0; i < 3; ++i) {
        dim3 grid((ROW_TILES + tiles_per_block - 1) / tiles_per_block, B_);  // 20 x 8
        chamfer_kernel<<<grid, dim3(256), 0, stream>>>(
            gt, rest[i], ws + d2off[i], out, npts[i], ntiles[i]);
    }
    for (int i = 0; i < 3; ++i) {
        int n = B_ * npts[i];
        d2_reduce_kernel<<<dim3((n + 255) / 256), dim3(256), 0, stream>>>(
            ws + d2off[i], out, n, 0.55f / (float)n);
    }
    for (int i = 0; i < 3; ++i) {
        int total = B_ * nedges[i];
        edge_kernel<<<dim3((total + 255) / 256), dim3(256), 0, stream>>>(
            pred[i], edges[i], out, npts[i], nedges[i], 1.0f / (float)total);
    }
    for (int i = 0; i < 3; ++i) {
        int total = B_ * npts[i];
        float w = lapc[i] / (float)total;
        lap_kernel<<<dim3((total + 255) / 256), dim3(256), 0, stream>>>(
            pred[i], before[i], lapi[i], out, npts[i], w, (i > 0) ? w : 0.0f);
    }
}